// Aggregator_86019605004680
// MI455X (gfx1250) — compile-verified
//
#include <hip/hip_runtime.h>

// Problem constants (fixed by the reference harness)
constexpr int BATCH  = 16;
constexpr int NCHUNK = 16;
constexpr int QLEN   = 64;
constexpr int DMODEL = 512;
constexpr int DOCL   = 512;   // DOC_MAX_LEN
constexpr int STEP   = 127;   // DOC_STRIDE - 1
constexpr int PLEN   = 2416;  // merged doc length
constexpr int D4     = DMODEL / 4;          // 128 float4 per row
constexpr int QN     = QLEN * DMODEL;       // 32768 elements per (b, chunk) question slab

typedef float v2f __attribute__((ext_vector_type(2)));
typedef float v4f __attribute__((ext_vector_type(4)));
typedef float v8f __attribute__((ext_vector_type(8)));

// ---------------------------------------------------------------------------
// Question merge: out_q[b][n] = (1/16) * sum_{c=0..15} x_q[b*16+c][n]
// Done as D = A x B with A = ones(16x4)/16 via V_WMMA_F32_16X16X4_F32,
// chained 4x (K=4 per step) to cover all 16 chunks. All D rows are identical,
// lanes 0..15 of D vgpr0 hold the 16 column results.
// One chain (16 output columns) per wave. 32768 waves total, all lanes live.
// ---------------------------------------------------------------------------
__global__ __launch_bounds__(256) void q_mean_wmma(const float* __restrict__ xq,
                                                   float* __restrict__ outq) {
    const int lane = threadIdx.x & 31;
    const int wave = blockIdx.x * 8 + (threadIdx.x >> 5); // 0 .. 32767
    const int b    = wave >> 11;        // wave / 2048
    const int tile = wave & 2047;       // 2048 tiles of 16 columns per doc
    const int n0   = tile << 4;
    const int col  = n0 + (lane & 15);
    const int csel = (lane >> 4) << 1;  // lanes 0-15 -> chunks {c0,c0+1}, 16-31 -> {c0+2,c0+3}

    const float* base = xq + (size_t)b * NCHUNK * QN + col;

    const v2f a = {0.0625f, 0.0625f};   // (1/16) * ones
    v8f acc = {};
#pragma unroll
    for (int s = 0; s < 4; ++s) {
        const int c0 = s * 4 + csel;
        v2f bv;
        bv.x = __builtin_nontemporal_load(base + (size_t)(c0    ) * QN);
        bv.y = __builtin_nontemporal_load(base + (size_t)(c0 + 1) * QN);
        // 8 args: (neg_a, A, neg_b, B, c_mod, C, reuse_a, reuse_b)
        acc = __builtin_amdgcn_wmma_f32_16x16x4_f32(false, a, false, bv,
                                                    (short)0, acc, false, false);
    }
    if (lane < 16) {
        __builtin_nontemporal_store(acc[0], outq + (size_t)b * QN + n0 + lane);
    }
}

// ---------------------------------------------------------------------------
// Document overlap-add as a gather:
//   out_d[b][p][:] = (1/cnt) * sum_{c=c_lo..c_hi} x_d[b*16+c][p - c*127][:]
//   c_hi = min(15, p/127); c_lo = max(0, ceil((p-510)/127)); cnt >= 1 always.
// Fixed-depth-5 masked unroll (interior rows have 4-5 real contributors) for
// full memory-level parallelism. 128-bit per thread, fully coalesced,
// non-temporal (streaming, zero reuse).
// ---------------------------------------------------------------------------
__global__ __launch_bounds__(256) void doc_overlap_add(const v4f* __restrict__ xd,
                                                       v4f* __restrict__ outd) {
    const int half = threadIdx.x >> 7;        // two rows per 256-thread block
    const int d4   = threadIdx.x & 127;
    const int row  = blockIdx.x * 2 + half;   // b * PLEN + p
    const int b    = row / PLEN;
    const int p    = row - b * PLEN;

    int c_hi = p / STEP;              if (c_hi > NCHUNK - 1) c_hi = NCHUNK - 1;
    int c_lo = (p > DOCL - 2) ? (p - (DOCL - 2) + STEP - 1) / STEP : 0;
    const float inv = 1.0f / (float)(c_hi - c_lo + 1);

    const v4f* base = xd + (size_t)b * NCHUNK * DOCL * D4 + d4;

    v4f acc = {0.f, 0.f, 0.f, 0.f};
#pragma unroll
    for (int j = 0; j < 5; ++j) {
        const int cj = c_lo + j;
        const int c  = (cj <= c_hi) ? cj : c_hi;     // clamp (duplicate hits L0)
        const float w = (cj <= c_hi) ? inv : 0.0f;   // mask + fold normalization
        const int i  = p - c * STEP;                 // token index within chunk
        const v4f v = __builtin_nontemporal_load(
            base + (size_t)c * DOCL * D4 + (size_t)i * D4);
        acc.x = fmaf(w, v.x, acc.x);
        acc.y = fmaf(w, v.y, acc.y);
        acc.z = fmaf(w, v.z, acc.z);
        acc.w = fmaf(w, v.w, acc.w);
    }
    __builtin_nontemporal_store(acc, outd + (size_t)row * D4 + d4);
}

extern "C" void kernel_launch(void* const* d_in, const int* in_sizes, int n_in,
                              void* d_out, int out_size, void* d_ws, size_t ws_size,
                              hipStream_t stream) {
    (void)in_sizes; (void)n_in; (void)d_ws; (void)ws_size; (void)out_size;

    const float* xq = (const float*)d_in[0];   // (256, 64, 512) fp32
    const float* xd = (const float*)d_in[1];   // (256, 512, 512) fp32

    float* out   = (float*)d_out;
    float* out_q = out;                                     // (16, 64, 512)
    float* out_d = out + (size_t)BATCH * QLEN * DMODEL;     // (16, 2416, 512)

    // 32768 waves -> 4096 blocks of 8 waves
    q_mean_wmma<<<4096, 256, 0, stream>>>(xq, out_q);

    // 16*2416 rows, 2 rows per block
    doc_overlap_add<<<(BATCH * PLEN) / 2, 256, 0, stream>>>(
        (const v4f*)xd, (v4f*)out_d);
}